// MyRWRModel_53120155516950
// MI455X (gfx1250) — compile-verified
//
#include <hip/hip_runtime.h>
#include <cstdint>
#include <cstddef>

// ---------------------------------------------------------------------------
// RWR (random-walk-with-restart) fused kernel for MI455X (gfx1250, wave32).
//
// Roofline: ~12 MFLOP total, ~4.8 MB total traffic -> ~0.2us of HBM time at
// 23.3 TB/s. Entirely launch-latency bound => one fused kernel, all state in
// LDS/registers. WMMA is deliberately NOT used: the reference "matmuls" are
// one-hot gathers and the stencil is a sparse K=8 GEMV with per-sample
// matrices — matrix cores cannot help. The CDNA5-specific path used instead
// is the async global->LDS copy (GLOBAL_LOAD_ASYNC_TO_LDS_B32 + ASYNCcnt).
// ---------------------------------------------------------------------------

namespace {
constexpr int kH = 32;
constexpr int kW = 32;
constexpr int kP = 12;
constexpr int kPQ = 24;          // P + Q
constexpr int kSteps = 12;       // K_STEP
constexpr int kCells = kH * kW;  // 1024
constexpr int kTeDepth = 24 * 7; // 168
constexpr int kPadW = kW + 2;    // 34
constexpr int kPadN = (kH + 2) * kPadW; // 1156
constexpr int kThreads = kCells; // 1024 threads = 32 waves (wave32)
}

typedef __attribute__((address_space(1))) int gint;
typedef __attribute__((address_space(3))) int lint;

__device__ __forceinline__ void async_global_to_lds_b32(const float* gsrc,
                                                        float* lds_dst) {
#if __has_builtin(__builtin_amdgcn_global_load_async_to_lds_b32)
  __builtin_amdgcn_global_load_async_to_lds_b32(
      (gint*)(uintptr_t)gsrc, (lint*)lds_dst, /*offset=*/0, /*cpol=*/0);
#else
  uint32_t lds_off = (uint32_t)(uintptr_t)(lint*)lds_dst;
  uint64_t ga = (uint64_t)(uintptr_t)gsrc;
  asm volatile("global_load_async_to_lds_b32 %0, %1, off"
               :: "v"(lds_off), "v"(ga) : "memory");
#endif
}

__device__ __forceinline__ void wait_asynccnt_zero() {
#if __has_builtin(__builtin_amdgcn_s_wait_asynccnt)
  __builtin_amdgcn_s_wait_asynccnt(0);
#else
  asm volatile("s_wait_asynccnt 0x0" ::: "memory");
#endif
}

__global__ __launch_bounds__(kThreads) void rwr_fused_kernel(
    const float* __restrict__ X,   // (B, P, 32, 32, 1)
    const int* __restrict__ TE,    // (B, 24, 2)
    const float* __restrict__ Wt,  // (1192, 8)
    const float* __restrict__ Wr,  // (1192, 1)
    const float* __restrict__ Wb,  // (1192, 1)
    float* __restrict__ outY,      // (B, 1, 32, 32, 1)
    float* __restrict__ outX,      // (B, P, 32, 32, 1)
    float* __restrict__ outTP,     // (B, P, 32, 32, 8, 1)
    float* __restrict__ outR,      // (B, P, 32, 32, 1)
    float* __restrict__ outRes)    // (B, P, 32, 32, 1, 13)
{
  __shared__ float s_tp[kCells * 8];   // 32 KB: trans_prob per cell
  __shared__ float s_x[2][kPadN];      // double-buffered padded grid

  const int blk = blockIdx.x;          // b * P + p
  const int b = blk / kP;
  const int p = blk - b * kP;
  const int tid = threadIdx.x;         // cell = h*32 + w
  const int h = tid >> 5;
  const int w = tid & (kW - 1);
  const int padIdx = (h + 1) * kPadW + (w + 1);

  // Zero buffer1 fully and buffer0's pad ring only (buffer0 interior is
  // filled by the async copy; avoid LDS-write vs async-write races).
  for (int i = tid; i < kPadN; i += kThreads) {
    const int r0 = i / kPadW;
    const int c0 = i - r0 * kPadW;
    s_x[1][i] = 0.0f;
    if (r0 == 0 || r0 == kH + 1 || c0 == 0 || c0 == kW + 1) s_x[0][i] = 0.0f;
  }

  // CDNA5 async copy: stage this (b,p)'s 32x32 X tile straight into the
  // padded LDS buffer, one b32 per lane, tracked by ASYNCcnt.
  async_global_to_lds_b32(X + (size_t)blk * kCells + tid, &s_x[0][padIdx]);

  // Time-embedding index (uniform per block): day*24 + hour.
  const int teBase = (b * kPQ + p) * 2;
  const int te = TE[teBase] * 24 + TE[teBase + 1];
  const int cellRow = kTeDepth + tid;

  // One-hot matmul == row gather: logits, restart, bias.
  const float4* Wt4 = reinterpret_cast<const float4*>(Wt);
  const float4 ta = Wt4[te * 2 + 0];
  const float4 tb = Wt4[te * 2 + 1];
  const float4 ca = Wt4[cellRow * 2 + 0];
  const float4 cb = Wt4[cellRow * 2 + 1];
  float logit[8] = {ta.x + ca.x, ta.y + ca.y, ta.z + ca.z, ta.w + ca.w,
                    tb.x + cb.x, tb.y + cb.y, tb.z + cb.z, tb.w + cb.w};

  float m = logit[0];
#pragma unroll
  for (int d = 1; d < 8; ++d) m = fmaxf(m, logit[d]);
  float tpl[8];
  float ssum = 0.0f;
#pragma unroll
  for (int d = 0; d < 8; ++d) { tpl[d] = expf(logit[d] - m); ssum += tpl[d]; }
  const float inv = 1.0f / ssum;
#pragma unroll
  for (int d = 0; d < 8; ++d) { tpl[d] *= inv; s_tp[tid * 8 + d] = tpl[d]; }

  const float rs = 1.0f / (1.0f + expf(-(Wr[te] + Wr[cellRow])));
  const float biasW = Wb[te] + Wb[cellRow];

  // trans_prob and restart outputs (32B-aligned float4 stores for tp).
  const size_t cellG = (size_t)blk * kCells + tid;
  float4* tpOut = reinterpret_cast<float4*>(outTP + cellG * 8);
  tpOut[0] = make_float4(tpl[0], tpl[1], tpl[2], tpl[3]);
  tpOut[1] = make_float4(tpl[4], tpl[5], tpl[6], tpl[7]);
  outR[cellG] = rs;

  wait_asynccnt_zero();  // per-wave: X tile resident in LDS
  __syncthreads();       // workgroup: tp + x tile visible to all waves

  // Pre-gather the 8 *neighbor* weights (loop-invariant):
  //   x_trans[c] = sum_d tp[nbr_d(c), d] * x[nbr_d(c)],
  // DIRS order d=0..7 -> offsets (-1,-1)(-1,0)(-1,1)(0,-1)(0,1)(1,-1)(1,0)(1,1)
  const int drs[8] = {-1, -1, -1, 0, 0, 1, 1, 1};
  const int dcs[8] = {-1, 0, 1, -1, 1, -1, 0, 1};
  float wgt[8];
#pragma unroll
  for (int d = 0; d < 8; ++d) {
    const int nh = h + drs[d];
    const int nw = w + dcs[d];
    const bool ok = ((unsigned)nh < (unsigned)kH) && ((unsigned)nw < (unsigned)kW);
    wgt[d] = ok ? s_tp[(nh * kW + nw) * 8 + d] : 0.0f;
  }
  const int poff[8] = {-kPadW - 1, -kPadW, -kPadW + 1, -1, 1,
                       kPadW - 1, kPadW, kPadW + 1};

  float x = s_x[0][padIdx];       // initial state from the async-staged tile
  const float xb = biasW * x;     // X_bias
  float res[kSteps + 1];
  res[0] = x;

  // 12 stencil steps; double buffering => one barrier per step.
#pragma unroll
  for (int k = 1; k <= kSteps; ++k) {
    float* cur = s_x[k & 1];
    cur[padIdx] = x;
    __syncthreads();
    float acc = wgt[0] * cur[padIdx + poff[0]];
#pragma unroll
    for (int d = 1; d < 8; ++d) acc = fmaf(wgt[d], cur[padIdx + poff[d]], acc);
    x = (1.0f - rs) * acc + rs * x + xb;
    res[k] = x;
  }

  // Final outputs.
  outX[cellG] = x;
  float* ro = outRes + cellG * 13;
#pragma unroll
  for (int k = 0; k <= kSteps; ++k) ro[k] = res[k];
  if (p == kP - 1) outY[(size_t)b * kCells + tid] = x;
}

extern "C" void kernel_launch(void* const* d_in, const int* in_sizes, int n_in,
                              void* d_out, int out_size, void* d_ws, size_t ws_size,
                              hipStream_t stream) {
  (void)n_in; (void)out_size; (void)d_ws; (void)ws_size;

  const float* X  = (const float*)d_in[0];
  const int*   TE = (const int*)d_in[1];
  const float* Wt = (const float*)d_in[2];
  const float* Wr = (const float*)d_in[3];
  const float* Wb = (const float*)d_in[4];

  const int B = in_sizes[0] / (kP * kCells);  // X is (B, P, 32, 32, 1)

  float* out    = (float*)d_out;
  float* outY   = out;                               // B*1024
  float* outX   = outY + (size_t)B * kCells;         // B*12*1024
  float* outTP  = outX + (size_t)B * kP * kCells;    // B*12*1024*8
  float* outR   = outTP + (size_t)B * kP * kCells * 8;  // B*12*1024
  float* outRes = outR + (size_t)B * kP * kCells;    // B*12*1024*13

  rwr_fused_kernel<<<B * kP, kThreads, 0, stream>>>(
      X, TE, Wt, Wr, Wb, outY, outX, outTP, outR, outRes);
}